// TopKCompressor_33440615367098
// MI455X (gfx1250) — compile-verified
//
#include <hip/hip_runtime.h>
#include <stdint.h>

#define HIST_BINS 65536
#define SEL_THREADS 1024
#define BINS_PER_THREAD (HIST_BINS / SEL_THREADS)

// ---------------------------------------------------------------------------
// Helpers
// ---------------------------------------------------------------------------
__device__ __forceinline__ uint32_t tk_abs_bits(float f) {
  return __float_as_uint(f) & 0x7fffffffu;
}

// Convert a generic pointer to a __shared__ object into a raw LDS byte offset
// (AS3 pointers are 32-bit on AMDGPU; addrspacecast generic->AS3 recovers it).
__device__ __forceinline__ uint32_t tk_lds_off(const void* p) {
  return (uint32_t)(uintptr_t)(__attribute__((address_space(3))) const char*)p;
}

__device__ __forceinline__ void tk_async_b128(uint32_t lds, unsigned long long gaddr) {
  // CDNA5 async global -> LDS copy (16B per lane), tracked with ASYNCcnt.
  asm volatile("global_load_async_to_lds_b128 %0, %1, off"
               :: "v"(lds), "v"(gaddr)
               : "memory");
}

// ---------------------------------------------------------------------------
// Kernel 0: zero the workspace (ctrl + both histograms)
// ---------------------------------------------------------------------------
extern "C" __global__ void tk_zero(uint32_t* __restrict__ p, int n) {
  int i = blockIdx.x * blockDim.x + threadIdx.x;
  if (i < n) p[i] = 0u;
}

// ---------------------------------------------------------------------------
// Kernel 1: histogram of top 16 bits of |x| bit pattern.
// One workgroup holds the full 65536-bin histogram in LDS (256KB of the
// 320KB/WGP CDNA5 LDS), merged once at the end with global atomics.
// ---------------------------------------------------------------------------
extern "C" __global__ void __launch_bounds__(1024)
tk_hist_hi(const float* __restrict__ x, long long n, uint32_t* __restrict__ hist) {
  extern __shared__ uint32_t bins[];
  for (int i = threadIdx.x; i < HIST_BINS; i += blockDim.x) bins[i] = 0u;
  __syncthreads();

  const long long n4     = n >> 2;
  const float4*   x4     = (const float4*)x;
  const long long stride = (long long)gridDim.x * blockDim.x;
  long long       i      = (long long)blockIdx.x * blockDim.x + threadIdx.x;

  for (; i < n4; i += stride) {
    __builtin_prefetch(x4 + i + 2 * stride, 0, 0);  // global_prefetch_b8
    float4 v = x4[i];
    atomicAdd(&bins[tk_abs_bits(v.x) >> 16], 1u);
    atomicAdd(&bins[tk_abs_bits(v.y) >> 16], 1u);
    atomicAdd(&bins[tk_abs_bits(v.z) >> 16], 1u);
    atomicAdd(&bins[tk_abs_bits(v.w) >> 16], 1u);
  }
  // scalar tail (n not divisible by 4)
  for (long long j = (n4 << 2) + (long long)blockIdx.x * blockDim.x + threadIdx.x;
       j < n; j += stride)
    atomicAdd(&bins[tk_abs_bits(x[j]) >> 16], 1u);

  __syncthreads();
  for (int b = threadIdx.x; b < HIST_BINS; b += blockDim.x) {
    uint32_t c = bins[b];
    if (c) atomicAdd(&hist[b], c);
  }
}

// ---------------------------------------------------------------------------
// Kernel 2/4: radix-select within a 65536-bin histogram, counting from the
// TOP (largest magnitudes). pass==0: write bucket + remaining-k to ctrl[0..1].
// pass==1: write final 32-bit bit-threshold to ctrl[3].
// ---------------------------------------------------------------------------
extern "C" __global__ void __launch_bounds__(SEL_THREADS)
tk_select(const uint32_t* __restrict__ hist, uint32_t* __restrict__ ctrl,
          int pass, unsigned long long k_in) {
  __shared__ uint32_t partial[SEL_THREADS];
  __shared__ unsigned long long kWant;
  if (threadIdx.x == 0)
    kWant = (pass == 0) ? k_in : (unsigned long long)ctrl[1];
  __syncthreads();

  uint32_t s = 0;
  const int base = threadIdx.x * BINS_PER_THREAD;
  for (int j = 0; j < BINS_PER_THREAD; ++j) s += hist[base + j];
  partial[threadIdx.x] = s;
  __syncthreads();

  if (threadIdx.x == 0) {
    const unsigned long long k = kWant;
    unsigned long long acc = 0;
    int t = SEL_THREADS - 1;
    for (; t > 0; --t) {
      if (acc + (unsigned long long)partial[t] >= k) break;
      acc += partial[t];
    }
    int bsel = t * BINS_PER_THREAD;
    for (int j = BINS_PER_THREAD - 1; j >= 0; --j) {
      uint32_t c = hist[t * BINS_PER_THREAD + j];
      if (acc + (unsigned long long)c >= k) { bsel = t * BINS_PER_THREAD + j; break; }
      acc += c;
    }
    if (pass == 0) {
      ctrl[0] = (uint32_t)bsel;                       // threshold top-16 bucket
      ctrl[1] = (uint32_t)(k - acc);                  // still needed inside it
    } else {
      ctrl[3] = (ctrl[0] << 16) | (uint32_t)bsel;     // full 32-bit threshold
    }
  }
}

// ---------------------------------------------------------------------------
// Kernel 3: histogram of low 16 bits for elements inside the threshold bucket.
// ---------------------------------------------------------------------------
extern "C" __global__ void __launch_bounds__(1024)
tk_hist_lo(const float* __restrict__ x, long long n,
           const uint32_t* __restrict__ ctrl, uint32_t* __restrict__ hist) {
  extern __shared__ uint32_t bins[];
  for (int i = threadIdx.x; i < HIST_BINS; i += blockDim.x) bins[i] = 0u;
  __syncthreads();

  const uint32_t  hb     = ctrl[0];
  const long long n4     = n >> 2;
  const float4*   x4     = (const float4*)x;
  const long long stride = (long long)gridDim.x * blockDim.x;
  long long       i      = (long long)blockIdx.x * blockDim.x + threadIdx.x;

  for (; i < n4; i += stride) {
    float4 v = x4[i];
    uint32_t a;
    a = tk_abs_bits(v.x); if ((a >> 16) == hb) atomicAdd(&bins[a & 0xffffu], 1u);
    a = tk_abs_bits(v.y); if ((a >> 16) == hb) atomicAdd(&bins[a & 0xffffu], 1u);
    a = tk_abs_bits(v.z); if ((a >> 16) == hb) atomicAdd(&bins[a & 0xffffu], 1u);
    a = tk_abs_bits(v.w); if ((a >> 16) == hb) atomicAdd(&bins[a & 0xffffu], 1u);
  }
  for (long long j = (n4 << 2) + (long long)blockIdx.x * blockDim.x + threadIdx.x;
       j < n; j += stride) {
    uint32_t a = tk_abs_bits(x[j]);
    if ((a >> 16) == hb) atomicAdd(&bins[a & 0xffffu], 1u);
  }

  __syncthreads();
  for (int b = threadIdx.x; b < HIST_BINS; b += blockDim.x) {
    uint32_t c = bins[b];
    if (c) atomicAdd(&hist[b], c);
  }
}

// ---------------------------------------------------------------------------
// Kernel 5: apply threshold. Streams input through LDS via CDNA5 async
// global->LDS copies, double-buffered with s_wait_asynccnt, and writes the
// masked result. Each lane reads back only its own staged 16B, so only the
// issuing wave's ASYNCcnt matters (no cross-wave barrier needed).
// ---------------------------------------------------------------------------
extern "C" __global__ void __launch_bounds__(256)
tk_apply(const float* __restrict__ x, float* __restrict__ out, long long n,
         const uint32_t* __restrict__ ctrl, int iters) {
  extern __shared__ char smem[];
  float4* buf = (float4*)smem;  // 2 * blockDim.x float4 slots

  const uint32_t  T      = ctrl[3];
  const float4*   x4     = (const float4*)x;
  float4*         o4     = (float4*)out;
  const long long n4     = n >> 2;
  const long long stride = (long long)gridDim.x * blockDim.x;
  const long long i0     = (long long)blockIdx.x * blockDim.x + threadIdx.x;

  const uint32_t lds0 = tk_lds_off(&buf[threadIdx.x]);
  const uint32_t lds1 = tk_lds_off(&buf[blockDim.x + threadIdx.x]);

  // Prologue: fill the two pipeline slots.
  if (i0 < n4)
    tk_async_b128(lds0, (unsigned long long)(uintptr_t)(x4 + i0));
  if (iters > 1 && i0 + stride < n4)
    tk_async_b128(lds1, (unsigned long long)(uintptr_t)(x4 + i0 + stride));

  for (int it = 0; it < iters; ++it) {
    const long long cur = i0 + (long long)it * stride;

    if (it == iters - 1)
      asm volatile("s_wait_asynccnt 0x0" ::: "memory");
    else
      asm volatile("s_wait_asynccnt 0x1" ::: "memory");

    if (cur < n4) {
      const int slot_base = (it & 1) ? (int)blockDim.x : 0;
      float4 v = buf[slot_base + threadIdx.x];
      float4 r;
      r.x = (tk_abs_bits(v.x) >= T) ? v.x : 0.0f;
      r.y = (tk_abs_bits(v.y) >= T) ? v.y : 0.0f;
      r.z = (tk_abs_bits(v.z) >= T) ? v.z : 0.0f;
      r.w = (tk_abs_bits(v.w) >= T) ? v.w : 0.0f;
      o4[cur] = r;
    }

    const long long nxt = i0 + (long long)(it + 2) * stride;
    if (it + 2 < iters && nxt < n4) {
      // Ensure the ds_load from this slot finished before the async engine
      // overwrites it (DS vs async-LDS ordering is not guaranteed).
      asm volatile("s_wait_dscnt 0x0" ::: "memory");
      const uint32_t l = (it & 1) ? lds1 : lds0;
      tk_async_b128(l, (unsigned long long)(uintptr_t)(x4 + nxt));
    }
  }

  // scalar tail
  for (long long j = (n4 << 2) + i0; j < n; j += stride) {
    float v = x[j];
    out[j] = (tk_abs_bits(v) >= T) ? v : 0.0f;
  }
}

// ---------------------------------------------------------------------------
// Host launcher
// ---------------------------------------------------------------------------
extern "C" void kernel_launch(void* const* d_in, const int* in_sizes, int n_in,
                              void* d_out, int out_size, void* d_ws, size_t ws_size,
                              hipStream_t stream) {
  (void)n_in; (void)out_size; (void)ws_size;

  const float* x   = (const float*)d_in[0];
  float*       out = (float*)d_out;
  const long long n = (long long)in_sizes[0];

  // k = max(1, int(n * 0.01)) -- mirror Python truncation semantics.
  long long k = (long long)((double)n * 0.01);
  if (k < 1) k = 1;

  uint32_t* ctrl  = (uint32_t*)d_ws;        // 16 u32 control words
  uint32_t* histA = ctrl + 16;              // 65536 u32
  uint32_t* histB = histA + HIST_BINS;      // 65536 u32

  // 0) zero workspace (deterministic per call; ws is not re-poisoned between
  //    graph replays, so we must reset it ourselves).
  const int zn = 16 + 2 * HIST_BINS;
  tk_zero<<<(zn + 1023) / 1024, 1024, 0, stream>>>((uint32_t*)d_ws, zn);

  const size_t hist_lds = (size_t)HIST_BINS * sizeof(uint32_t);  // 256 KB / WGP

  // 1) coarse histogram (top 16 bits of |x| bit pattern)
  tk_hist_hi<<<256, 1024, hist_lds, stream>>>(x, n, histA);

  // 2) select coarse bucket + remaining count
  tk_select<<<1, SEL_THREADS, 0, stream>>>(histA, ctrl, 0, (unsigned long long)k);

  // 3) fine histogram (low 16 bits, filtered to the threshold bucket)
  tk_hist_lo<<<256, 1024, hist_lds, stream>>>(x, n, ctrl, histB);

  // 4) select fine bucket -> exact 32-bit threshold in ctrl[3]
  tk_select<<<1, SEL_THREADS, 0, stream>>>(histB, ctrl, 1, 0ull);

  // 5) apply threshold with async global->LDS double-buffered streaming
  const int block = 256;
  const int grid  = 2048;
  const long long n4  = n >> 2;
  const long long tot = (long long)grid * block;
  int iters = (int)((n4 + tot - 1) / tot);
  if (iters < 1) iters = 1;
  tk_apply<<<grid, block, (size_t)(2 * block) * sizeof(float4), stream>>>(
      x, out, n, ctrl, iters);
}